// GAT_Net_21784074125387
// MI455X (gfx1250) — compile-verified
//
#include <hip/hip_runtime.h>
#include <hip/hip_bf16.h>
#include <math.h>

// ---------------------------------------------------------------------------
// 2-layer GAT (PyG GATConv semantics) for MI455X / gfx1250, wave32.
// GEMMs run on the matrix engine via V_WMMA_F32_16X16X4_F32 (exact fp32).
// Edge phases are bandwidth-bound scatter/gather with f32 global atomics.
// ---------------------------------------------------------------------------

#define F_IN   256
#define HH     128   // HEADS*HID
#define HEADS  4
#define HID    32
#define CLS    32

typedef __attribute__((ext_vector_type(2))) float v2f;
typedef __attribute__((ext_vector_type(8))) float v8f;

// ---- monotone float<->uint mapping for atomic max on signed floats --------
__device__ __forceinline__ unsigned f2ord(float f) {
    unsigned u = __float_as_uint(f);
    return (u & 0x80000000u) ? ~u : (u | 0x80000000u);
}
__device__ __forceinline__ float ord2f(unsigned u) {
    return (u & 0x80000000u) ? __uint_as_float(u & 0x7fffffffu)
                             : __uint_as_float(~u);
}

// ---------------------------------------------------------------------------
__global__ void zero_fill(float* __restrict__ p, long long n) {
    long long t = (long long)blockIdx.x * blockDim.x + threadIdx.x;
    if (t < n) p[t] = 0.0f;
}

// ---- GEMM1: h1[N,128] = x[N,256] @ W1^T  (WMMA f32 16x16x4) ---------------
// One wave per 16-row tile; 8 column tiles (128 cols) accumulated per wave.
__global__ void gemm1_wmma(const float* __restrict__ x,
                           const float* __restrict__ W1,
                           float* __restrict__ h1, int n_nodes) {
    const int lane = threadIdx.x & 31;
    const int wave = threadIdx.x >> 5;
    const int tile = blockIdx.x * (blockDim.x >> 5) + wave;
    const int m0 = tile * 16;
    if (m0 >= n_nodes) return;                  // wave-uniform: EXEC stays all-1s

    const int r = lane & 15;                    // row within tile / col within tile
    const int kh = (lane >> 4) << 1;            // k sub-offset: 0 or 2

    v8f acc[8] = {};
    const float* xrow = x + (size_t)(m0 + r) * F_IN + kh;

    for (int k0 = 0; k0 < F_IN; k0 += 4) {
        __builtin_prefetch(xrow + k0 + 64, 0, 1);   // global_prefetch_b8
        v2f a;
        a.x = xrow[k0];
        a.y = xrow[k0 + 1];
#pragma unroll
        for (int nt = 0; nt < 8; nt++) {
            const float* wrow = W1 + (size_t)(nt * 16 + r) * F_IN + k0 + kh;
            v2f b;
            b.x = wrow[0];
            b.y = wrow[1];
            acc[nt] = __builtin_amdgcn_wmma_f32_16x16x4_f32(
                false, a, false, b, (short)0, acc[nt], false, false);
        }
    }
    const int rowbase = m0 + ((lane >> 4) << 3);    // +8 rows for upper half-wave
#pragma unroll
    for (int nt = 0; nt < 8; nt++)
#pragma unroll
        for (int v = 0; v < 8; v++)
            h1[(size_t)(rowbase + v) * HH + nt * 16 + r] = acc[nt][v];
}

// ---- GEMM2: z[N,32] = h[N,128] @ W2^T  (WMMA f32 16x16x4) -----------------
__global__ void gemm2_wmma(const float* __restrict__ h,
                           const float* __restrict__ W2,
                           float* __restrict__ z, int n_nodes) {
    const int lane = threadIdx.x & 31;
    const int wave = threadIdx.x >> 5;
    const int tile = blockIdx.x * (blockDim.x >> 5) + wave;
    const int m0 = tile * 16;
    if (m0 >= n_nodes) return;

    const int r = lane & 15;
    const int kh = (lane >> 4) << 1;

    v8f acc[2] = {};
    const float* hrow = h + (size_t)(m0 + r) * HH + kh;

    for (int k0 = 0; k0 < HH; k0 += 4) {
        v2f a;
        a.x = hrow[k0];
        a.y = hrow[k0 + 1];
#pragma unroll
        for (int nt = 0; nt < 2; nt++) {
            const float* wrow = W2 + (size_t)(nt * 16 + r) * HH + k0 + kh;
            v2f b;
            b.x = wrow[0];
            b.y = wrow[1];
            acc[nt] = __builtin_amdgcn_wmma_f32_16x16x4_f32(
                false, a, false, b, (short)0, acc[nt], false, false);
        }
    }
    const int rowbase = m0 + ((lane >> 4) << 3);
#pragma unroll
    for (int nt = 0; nt < 2; nt++)
#pragma unroll
        for (int v = 0; v < 8; v++)
            z[(size_t)(rowbase + v) * CLS + nt * 16 + r] = acc[nt][v];
}

// ---- attention coefficients, layer 1: one thread per (node, head) ---------
__global__ void attn_coef1(const float* __restrict__ h1,
                           const float* __restrict__ a_src,
                           const float* __restrict__ a_dst,
                           float* __restrict__ asc, float* __restrict__ adc,
                           int n_nodes) {
    int t = blockIdx.x * blockDim.x + threadIdx.x;
    if (t >= n_nodes * HEADS) return;
    int node = t >> 2, hd = t & 3;
    const float* hp = h1 + (size_t)node * HH + hd * HID;
    const float* as = a_src + hd * HID;
    const float* ad = a_dst + hd * HID;
    float s = 0.f, d = 0.f;
#pragma unroll
    for (int c = 0; c < HID; c++) {
        float v = hp[c];
        s += v * as[c];
        d += v * ad[c];
    }
    asc[t] = s;
    adc[t] = d;
}

// ---- attention coefficients, layer 2: one thread per node -----------------
__global__ void attn_coef2(const float* __restrict__ z,
                           const float* __restrict__ a_src,
                           const float* __restrict__ a_dst,
                           float* __restrict__ asc, float* __restrict__ adc,
                           int n_nodes) {
    int t = blockIdx.x * blockDim.x + threadIdx.x;
    if (t >= n_nodes) return;
    const float* zp = z + (size_t)t * CLS;
    float s = 0.f, d = 0.f;
#pragma unroll
    for (int c = 0; c < CLS; c++) {
        float v = zp[c];
        s += v * a_src[c];
        d += v * a_dst[c];
    }
    asc[t] = s;
    adc[t] = d;
}

// ---- edge pass A: logits + segment max (per dst, per head) ----------------
template <int H>
__global__ void edge_max(const int* __restrict__ ei, int E, int E2, int n_nodes,
                         const float* __restrict__ asc,
                         const float* __restrict__ adc,
                         float* __restrict__ elog, unsigned* __restrict__ mord) {
    int eid = blockIdx.x * blockDim.x + threadIdx.x;
    if (eid >= E2) return;
    int s, d;
    if (eid < E) { s = ei[eid]; d = ei[E + eid]; }
    else         { s = d = eid - E; }
#pragma unroll
    for (int hd = 0; hd < H; hd++) {
        float e = asc[s * H + hd] + adc[d * H + hd];
        e = (e > 0.f) ? e : 0.2f * e;            // leaky_relu(0.2)
        elog[(size_t)eid * H + hd] = e;
        atomicMax(&mord[d * H + hd], f2ord(e));
    }
}

// ---- edge pass B: p = exp(e - max) + segment sum ---------------------------
template <int H>
__global__ void edge_sum(const int* __restrict__ ei, int E, int E2,
                         const unsigned* __restrict__ mord,
                         float* __restrict__ elog, float* __restrict__ denom) {
    int eid = blockIdx.x * blockDim.x + threadIdx.x;
    if (eid >= E2) return;
    int d = (eid < E) ? ei[E + eid] : (eid - E);
#pragma unroll
    for (int hd = 0; hd < H; hd++) {
        float e = elog[(size_t)eid * H + hd];
        float p = expf(e - ord2f(mord[d * H + hd]));
        elog[(size_t)eid * H + hd] = p;
        atomicAdd(&denom[d * H + hd], p);
    }
}

// ---- edge pass C: alpha-weighted gather/scatter ---------------------------
// One thread per (edge, 4-channel chunk). CH = total channels, H = heads.
template <int H, int CH>
__global__ void edge_scatter(const int* __restrict__ ei, int E, int E2,
                             const float* __restrict__ hfeat,
                             const float* __restrict__ p,
                             const float* __restrict__ denom,
                             float* __restrict__ agg) {
    const int CHUNKS = CH / 4;
    long long t = (long long)blockIdx.x * blockDim.x + threadIdx.x;
    if (t >= (long long)E2 * CHUNKS) return;
    int eid = (int)(t / CHUNKS);
    int q   = (int)(t % CHUNKS);
    int s, d;
    if (eid < E) { s = ei[eid]; d = ei[E + eid]; }
    else         { s = d = eid - E; }
    int c0 = q * 4;
    int hd = c0 / (CH / H);
    float alpha = p[(size_t)eid * H + hd] / (denom[d * H + hd] + 1e-16f);
    const float4 hv = *(const float4*)(hfeat + (size_t)s * CH + c0);
    float* ap = agg + (size_t)d * CH + c0;
    atomicAdd(ap + 0, hv.x * alpha);
    atomicAdd(ap + 1, hv.y * alpha);
    atomicAdd(ap + 2, hv.z * alpha);
    atomicAdd(ap + 3, hv.w * alpha);
}

// ---- bias + ReLU (in-place on agg1) ---------------------------------------
__global__ void bias_relu(float* __restrict__ a, const float* __restrict__ b,
                          long long n) {
    long long t = (long long)blockIdx.x * blockDim.x + threadIdx.x;
    if (t >= n) return;
    float v = a[t] + b[t & (HH - 1)];
    a[t] = v > 0.f ? v : 0.f;
}

// ---- final: log_softmax(agg2 + b2) — one wave (32 lanes) per node ---------
__global__ void final_logsoftmax(const float* __restrict__ agg2,
                                 const float* __restrict__ b2,
                                 float* __restrict__ out, int n_nodes) {
    int lane = threadIdx.x & 31;
    int node = blockIdx.x * (blockDim.x >> 5) + (threadIdx.x >> 5);
    if (node >= n_nodes) return;
    float v = agg2[(size_t)node * CLS + lane] + b2[lane];
    float m = v;
#pragma unroll
    for (int off = 16; off >= 1; off >>= 1)
        m = fmaxf(m, __shfl_xor(m, off));
    float ex = expf(v - m);
#pragma unroll
    for (int off = 16; off >= 1; off >>= 1)
        ex += __shfl_xor(ex, off);
    out[(size_t)node * CLS + lane] = (v - m) - logf(ex);
}

// ---------------------------------------------------------------------------
extern "C" void kernel_launch(void* const* d_in, const int* in_sizes, int n_in,
                              void* d_out, int out_size, void* d_ws, size_t ws_size,
                              hipStream_t stream) {
    const float* x      = (const float*)d_in[0];
    const int*   ei     = (const int*)  d_in[1];
    const float* W1     = (const float*)d_in[2];
    const float* a_src1 = (const float*)d_in[3];
    const float* a_dst1 = (const float*)d_in[4];
    const float* b1     = (const float*)d_in[5];
    const float* W2     = (const float*)d_in[6];
    const float* a_src2 = (const float*)d_in[7];
    const float* a_dst2 = (const float*)d_in[8];
    const float* b2     = (const float*)d_in[9];
    float* out = (float*)d_out;

    const int N  = in_sizes[0] / F_IN;   // 50000
    const int E  = in_sizes[1] / 2;      // 800000
    const int E2 = E + N;                // + self loops

    // ---- workspace layout (floats) ----
    float* ws = (float*)d_ws;
    size_t o = 0;
    float*    h1    = ws + o; o += (size_t)N * HH;
    // zero-init region (accumulators + max sentinels, contiguous):
    size_t zbeg = o;
    float*    agg1  = ws + o; o += (size_t)N * HH;
    unsigned* m1    = (unsigned*)(ws + o); o += (size_t)N * HEADS;
    float*    den1  = ws + o; o += (size_t)N * HEADS;
    float*    agg2  = ws + o; o += (size_t)N * CLS;
    unsigned* m2    = (unsigned*)(ws + o); o += (size_t)N;
    float*    den2  = ws + o; o += (size_t)N;
    size_t zcount = o - zbeg;
    float*    asc1  = ws + o; o += (size_t)N * HEADS;
    float*    adc1  = ws + o; o += (size_t)N * HEADS;
    float*    e1    = ws + o; o += (size_t)E2 * HEADS;
    float*    z2    = ws + o; o += (size_t)N * CLS;
    float*    asc2  = ws + o; o += (size_t)N;
    float*    adc2  = ws + o; o += (size_t)N;
    float*    e2    = ws + o; o += (size_t)E2;

    const int T = 256;
    // 0) zero accumulation state (every call: graph-replay deterministic)
    zero_fill<<<(int)((zcount + T - 1) / T), T, 0, stream>>>(ws + zbeg, (long long)zcount);

    // 1) h1 = x @ W1^T (WMMA), attention coefficients
    {
        int tiles = N / 16;               // N multiple of 16
        int wpb = 4;                      // 4 waves/block (64 acc VGPRs per wave)
        gemm1_wmma<<<(tiles + wpb - 1) / wpb, wpb * 32, 0, stream>>>(x, W1, h1, N);
    }
    attn_coef1<<<(N * HEADS + T - 1) / T, T, 0, stream>>>(h1, a_src1, a_dst1, asc1, adc1, N);

    // 2) layer-1 edge softmax + aggregation
    edge_max<HEADS><<<(E2 + T - 1) / T, T, 0, stream>>>(ei, E, E2, N, asc1, adc1, e1, m1);
    edge_sum<HEADS><<<(E2 + T - 1) / T, T, 0, stream>>>(ei, E, E2, m1, e1, den1);
    {
        long long work = (long long)E2 * (HH / 4);
        edge_scatter<HEADS, HH><<<(int)((work + T - 1) / T), T, 0, stream>>>(
            ei, E, E2, h1, e1, den1, agg1);
    }

    // 3) bias + ReLU (in place)
    bias_relu<<<(int)(((long long)N * HH + T - 1) / T), T, 0, stream>>>(agg1, b1, (long long)N * HH);

    // 4) z2 = h @ W2^T (WMMA), layer-2 coefficients
    {
        int tiles = N / 16;
        int wpb = 8;
        gemm2_wmma<<<(tiles + wpb - 1) / wpb, wpb * 32, 0, stream>>>(agg1, W2, z2, N);
    }
    attn_coef2<<<(N + T - 1) / T, T, 0, stream>>>(z2, a_src2, a_dst2, asc2, adc2, N);

    // 5) layer-2 edge softmax + aggregation
    edge_max<1><<<(E2 + T - 1) / T, T, 0, stream>>>(ei, E, E2, N, asc2, adc2, e2, m2);
    edge_sum<1><<<(E2 + T - 1) / T, T, 0, stream>>>(ei, E, E2, m2, e2, den2);
    {
        long long work = (long long)E2 * (CLS / 4);
        edge_scatter<1, CLS><<<(int)((work + T - 1) / T), T, 0, stream>>>(
            ei, E, E2, z2, e2, den2, agg2);
    }

    // 6) out = log_softmax(agg2 + b2)
    final_logsoftmax<<<(N + 7) / 8, 8 * 32, 0, stream>>>(agg2, b2, out, N);
}